// SequenceWeightedCELoss2_91139206021300
// MI455X (gfx1250) — compile-verified
//
#include <hip/hip_runtime.h>
#include <stdint.h>

// Problem dims (fixed by the reference)
#define N_ 64
#define C_ 1024
#define L_ 2048

#define TL 256      // l-tile width per block (== blockDim.x)
#define CB 16       // c-rows per TDM chunk (16 * 256 * 4B = 16 KB)
#define NCHUNK (C_ / CB)

#define LOG2E 1.4426950408889634f
#define LN2   0.69314718055994531f

typedef unsigned int u32x4 __attribute__((ext_vector_type(4)));
typedef int          i32x4 __attribute__((ext_vector_type(4)));
typedef int          i32x8 __attribute__((ext_vector_type(8)));
typedef float        v2f   __attribute__((ext_vector_type(2)));
typedef float        v8f   __attribute__((ext_vector_type(8)));

// -----------------------------------------------------------------------------------
// CDNA5 Tensor Data Mover: DMA one 2D tile (CB rows x TL floats, row stride L_ elems)
// from global memory into LDS. Descriptor packing per CDNA5 ISA
// (D# group 0 / group 1 tables). Tracked with TENSORcnt.
// -----------------------------------------------------------------------------------
__device__ __forceinline__ void tdm_load_tile(uint32_t lds_addr, uint64_t gaddr) {
    u32x4 g0;
    g0.x = 1u;                                    // [1:0] count=1 (valid user D#)
    g0.y = lds_addr;                              // [63:32] LDS byte address
    g0.z = (uint32_t)gaddr;                       // [95:64] global_addr[31:0]
    g0.w = (uint32_t)((gaddr >> 32) & 0x01FFFFFFu) | (2u << 30); // addr[56:32] | type=2

    i32x8 g1;
    g1.s0 = 0x00020000;                           // [17:16] data_size=2 -> 4 bytes
    g1.s1 = (int)(256u << 16);                    // [79:48] tensor_dim0 = 256 (lo16)
    g1.s2 = (int)(1024u << 16);                   // dim0 hi=0 | [111:80] tensor_dim1=1024 (lo16)
    g1.s3 = (int)(256u << 16);                    // dim1 hi=0 | [127:112] tile_dim0 = 256
    g1.s4 = CB;                                   // [143:128] tile_dim1=16, tile_dim2=0
    g1.s5 = L_;                                   // [207:160] tensor_dim0_stride = 2048 (lo32)
    g1.s6 = 0;                                    // stride0 hi | stride1 lo
    g1.s7 = 0;

    i32x4 z4 = {0, 0, 0, 0};                      // groups 2/3: 2D tile, unused dims = 0
#if defined(__clang_major__) && __clang_major__ >= 23
    i32x8 z8 = {0, 0, 0, 0, 0, 0, 0, 0};
    __builtin_amdgcn_tensor_load_to_lds(g0, g1, z4, z4, z8, 0);
#else
    __builtin_amdgcn_tensor_load_to_lds(g0, g1, z4, z4, 0);
#endif
}

// -----------------------------------------------------------------------------------
// Kernel 1: logp[n,l] = logsumexp_c(inputs[n,:,l]) - inputs[n, tgt[n,l], l]
// One block per (l-tile, n). Streams C=1024 rows through a double-buffered LDS tile
// filled by TDM tensor_load_to_lds (wave 0 issues, TENSORcnt-paced, barrier-published);
// online one-pass softmax in base-2 (v_exp_f32-native).
// -----------------------------------------------------------------------------------
__global__ __launch_bounds__(256) void k_logp(const float* __restrict__ inp,
                                              const int* __restrict__ tgt,
                                              float* __restrict__ logp) {
    __shared__ float tile[2][CB * TL];   // 2 x 16 KB

    const int t  = threadIdx.x;          // 0..255  == l within tile
    const int n  = blockIdx.y;
    const int l0 = blockIdx.x * TL;
    const int l  = l0 + t;

    const int cstar = tgt[(size_t)n * L_ + l];

    const uint64_t gbase = (uint64_t)(uintptr_t)(inp + (size_t)n * C_ * L_ + l0);

    // Prime chunk 0 into buffer 0 (TDM engine does the strided 16 KB tile copy)
    if (t == 0)
        tdm_load_tile((uint32_t)(size_t)&tile[0][0], gbase);

    float m = -3.0e38f;   // running max of x*log2(e)
    float s = 0.0f;       // running sum of 2^(y - m)
    float tval = 0.0f;    // inputs[n, cstar, l]

    for (int ch = 0; ch < NCHUNK; ++ch) {
        const int c0  = ch * CB;
        const int cur = ch & 1;

        if (t == 0) {
            if (ch + 1 < NCHUNK) {
                // Queue next chunk into the other buffer, then wait for current only
                tdm_load_tile((uint32_t)(size_t)&tile[cur ^ 1][0],
                              gbase + (uint64_t)(c0 + CB) * (L_ * 4));
                __builtin_amdgcn_s_wait_tensorcnt((short)1);
            } else {
                __builtin_amdgcn_s_wait_tensorcnt((short)0);
            }
        }
        __syncthreads();     // publish TDM-written LDS to all 8 waves

        // Online logsumexp over the 16 resident rows (common path = 1 exp2/element)
        #pragma unroll
        for (int r = 0; r < CB; ++r) {
            const float x = tile[cur][r * TL + t];
            const float y = x * LOG2E;
            if (y > m) {
                s = s * __builtin_amdgcn_exp2f(m - y) + 1.0f;
                m = y;
            } else {
                s += __builtin_amdgcn_exp2f(y - m);
            }
        }
        if (cstar >= c0 && cstar < c0 + CB)
            tval = tile[cur][(cstar - c0) * TL + t];

        __syncthreads();     // safe for TDM to overwrite buffer `cur` next iteration
    }

    const float lse = LN2 * (m + __builtin_amdgcn_logf(s));   // natural logsumexp
    logp[(size_t)n * L_ + l] = lse - tval;
}

// -----------------------------------------------------------------------------------
// Kernel 2: S[l] = sum_n logp[n,l]   (fixed order -> deterministic)
// -----------------------------------------------------------------------------------
__global__ __launch_bounds__(256) void k_colsum(const float* __restrict__ logp,
                                                float* __restrict__ S) {
    const int l = blockIdx.x * 256 + threadIdx.x;
    float acc = 0.0f;
    #pragma unroll 8
    for (int n = 0; n < N_; ++n) acc += logp[(size_t)n * L_ + l];
    S[l] = acc;
}

// -----------------------------------------------------------------------------------
// Kernel 3: WMMA epilogue.  wm[n,l] = weights[tgt[n,l], l];
//   dot[n] = sum_l wm*S,  wsum[n] = sum_l wm   computed as a 16x16x4 F32 WMMA:
//   A = 16 rows of wm (K=4 slice),  B col0 = S, col1 = ones, cols 2..15 = 0.
// K split over 64 blocks (32 l's each) for parallelism; 4 waves cover the 64 rows.
// Partials written per block, reduced in fixed order by k_final (deterministic).
// -----------------------------------------------------------------------------------
__global__ __launch_bounds__(128) void k_row_wmma(const int* __restrict__ tgt,
                                                  const float* __restrict__ w,
                                                  const float* __restrict__ S,
                                                  float* __restrict__ pd,
                                                  float* __restrict__ pw) {
    const int lane = threadIdx.x & 31;
    const int wave = threadIdx.x >> 5;   // 0..3 -> rows [16*wave, 16*wave+16)
    const int j    = lane & 15;          // A: row M; B: column N; D: column N
    const int kh   = lane >> 4;          // lanes 16-31 carry K = 2,3 of each step
    const int n    = wave * 16 + j;
    const int l0   = blockIdx.x * 32;    // this block's K slice

    const float bs = (j == 0) ? 1.0f : 0.0f;   // B col 0 <- S[l]
    const float bo = (j == 1) ? 1.0f : 0.0f;   // B col 1 <- 1.0

    v8f acc = {0.0f, 0.0f, 0.0f, 0.0f, 0.0f, 0.0f, 0.0f, 0.0f};

    #pragma unroll
    for (int k0 = 0; k0 < 32; k0 += 4) {
        const int la = l0 + k0 + 2 * kh;             // this lane's K pair
        const int c0 = tgt[n * L_ + la];
        const int c1 = tgt[n * L_ + la + 1];
        v2f A, B;
        A.x = w[(size_t)c0 * L_ + la];               // wm[n, la]
        A.y = w[(size_t)c1 * L_ + la + 1];           // wm[n, la+1]
        B.x = bs * S[la]     + bo;
        B.y = bs * S[la + 1] + bo;
        acc = __builtin_amdgcn_wmma_f32_16x16x4_f32(
                  false, A, false, B, (short)0, acc, false, false);
    }

    // D layout: lane j<16 holds N=j rows M=r (acc[r]); lane j+16 holds N=j rows M=r+8
    const int mb = kh * 8;
    if (j == 0) {
        #pragma unroll
        for (int r = 0; r < 8; ++r)
            pd[blockIdx.x * 64 + wave * 16 + mb + r] = acc[r];
    } else if (j == 1) {
        #pragma unroll
        for (int r = 0; r < 8; ++r)
            pw[blockIdx.x * 64 + wave * 16 + mb + r] = acc[r];
    }
}

// -----------------------------------------------------------------------------------
// Kernel 4: reduce K-partials (fixed order), resc[n] = dot/wsum, out = mean_n resc
// -----------------------------------------------------------------------------------
__global__ void k_final(const float* __restrict__ pd, const float* __restrict__ pw,
                        float* __restrict__ out) {
    __shared__ float r64[64];
    const int t = threadIdx.x;           // 64 threads
    if (t < 64) {
        float d = 0.0f, s = 0.0f;
        for (int b = 0; b < 64; ++b) { d += pd[b * 64 + t]; s += pw[b * 64 + t]; }
        r64[t] = d / s;
    }
    __syncthreads();
    if (t == 0) {
        float a = 0.0f;
        for (int nn = 0; nn < N_; ++nn) a += r64[nn];
        out[0] = a * (1.0f / N_);
    }
}

// -----------------------------------------------------------------------------------
extern "C" void kernel_launch(void* const* d_in, const int* in_sizes, int n_in,
                              void* d_out, int out_size, void* d_ws, size_t ws_size,
                              hipStream_t stream) {
    const float* inp = (const float*)d_in[0];  // (N, C, L) f32
    const int*   tgt = (const int*)  d_in[1];  // (N, L) integer targets
    const float* w   = (const float*)d_in[2];  // (C, L) f32

    float* ws   = (float*)d_ws;
    float* logp = ws;                          // N*L floats
    float* S    = ws + (size_t)N_ * L_;        // L floats
    float* pd   = S + L_;                      // 64*64 floats (dot partials)
    float* pw   = pd + 64 * 64;                // 64*64 floats (wsum partials)
    float* out  = (float*)d_out;

    dim3 g1(L_ / TL, N_);
    k_logp    <<<g1,           dim3(256), 0, stream>>>(inp, tgt, logp);
    k_colsum  <<<dim3(L_/256), dim3(256), 0, stream>>>(logp, S);
    k_row_wmma<<<dim3(64),     dim3(128), 0, stream>>>(tgt, w, S, pd, pw);
    k_final   <<<dim3(1),      dim3(64),  0, stream>>>(pd, pw, out);
}